// LSTMActor_75033078661399
// MI455X (gfx1250) — compile-verified
//
#include <hip/hip_runtime.h>
#include <hip/hip_bf16.h>
#include <math.h>

#define HIDDEN   256
#define IN_SIZE  32768
#define OUT_SIZE 50257

typedef __attribute__((ext_vector_type(2))) float v2f;
typedef __attribute__((ext_vector_type(8))) float v8f;

// ---------------------------------------------------------------------------
// y_partial[slice, m] = sum_{k in slice} A[m,k] * x[k]
// One wave (32 lanes) per (M-tile of 16 rows, K-slice of KT cols).
// Uses V_WMMA_F32_16X16X4_F32 (full fp32 precision, matches the reference).
//   A tile 16x4:  lane l (0..15) holds A[m=mbase+l][K=0,1] in VGPR0/1,
//                 lane l+16 holds K=2,3  -> one float2 per lane.
//   B tile 4x16:  x values broadcast into every column (same float2 per lane
//                 half), so every column of D equals the desired partial y.
//   D 16x16 f32:  column 0 lives in lanes 0 (M=0..7) and 16 (M=8..15).
// Pointers are hoisted and the body unrolled so loads use immediate offsets
// (no per-load 64-bit address math in the hot loop).
// ---------------------------------------------------------------------------
__global__ __launch_bounds__(32) void wmma_matvec_partial(
    const float* __restrict__ A, const float* __restrict__ x,
    float* __restrict__ partials, int M, int K, int KT)
{
    const int lane  = threadIdx.x;      // 0..31
    const int l     = lane & 15;
    const int hi    = lane >> 4;        // 0: K pair {0,1}; 1: K pair {2,3}
    const int mbase = blockIdx.x * 16;
    const size_t k0 = (size_t)blockIdx.y * (size_t)KT;

    const float* __restrict__ ap =
        A + (size_t)(mbase + l) * (size_t)K + k0 + 2 * hi;
    const float* __restrict__ xp = x + k0 + 2 * hi;

    v8f c = {};
    for (int k = 0; k < KT; k += 32) {          // 8 WMMAs per iteration
        #pragma unroll
        for (int u = 0; u < 32; u += 4) {
            v2f a = *(const v2f*)(ap + k + u);  // A[m][k..k+1] for this half
            v2f b = *(const v2f*)(xp + k + u);  // x[k..k+1] (broadcast per half)
            c = __builtin_amdgcn_wmma_f32_16x16x4_f32(
                    /*neg_a=*/false, a, /*neg_b=*/false, b,
                    /*c_mod=*/(short)0, c, /*reuse_a=*/false, /*reuse_b=*/false);
        }
    }

    if (l == 0) {  // lanes 0 and 16 carry column 0 of D
        float* out = partials + (size_t)blockIdx.y * (size_t)M + mbase + hi * 8;
        #pragma unroll
        for (int r = 0; r < 8; ++r) out[r] = c[r];
    }
}

// v[j] = PReLU( sum_slices partials[s][j] + in_b[j] )
__global__ void reduce_bias_prelu(const float* __restrict__ partials,
                                  const float* __restrict__ in_b,
                                  const float* __restrict__ prelu_w,
                                  float* __restrict__ v, int M, int S)
{
    int j = blockIdx.x * blockDim.x + threadIdx.x;
    if (j >= M) return;
    float acc = in_b[j];
    for (int s = 0; s < S; ++s) acc += partials[(size_t)s * M + j];
    float slope = prelu_w[0];
    v[j] = (acc >= 0.0f) ? acc : slope * acc;
}

// Combine gate partials, apply LSTM nonlinearities, emit h (ws + d_out) and c.
// Gate order (PyTorch): i, f, g, o at rows j, j+256, j+512, j+768.
__global__ __launch_bounds__(HIDDEN) void lstm_combine(
    const float* __restrict__ pg_ih, const float* __restrict__ pg_hh, int S,
    const float* __restrict__ b_ih, const float* __restrict__ b_hh,
    const float* __restrict__ c_prev,
    float* __restrict__ h_ws, float* __restrict__ d_out, int layer)
{
    int j = threadIdx.x;   // 0..255
    float g[4];
    #pragma unroll
    for (int q = 0; q < 4; ++q) {
        int row = q * HIDDEN + j;
        float acc = b_ih[row] + b_hh[row];
        for (int s = 0; s < S; ++s)
            acc += pg_ih[(size_t)s * (4 * HIDDEN) + row]
                 + pg_hh[(size_t)s * (4 * HIDDEN) + row];
        g[q] = acc;
    }
    float gi = 1.0f / (1.0f + expf(-g[0]));
    float gf = 1.0f / (1.0f + expf(-g[1]));
    float gg = tanhf(g[2]);
    float go = 1.0f / (1.0f + expf(-g[3]));
    float c_new = gf * c_prev[j] + gi * gg;
    float h_new = go * tanhf(c_new);

    h_ws[j] = h_new;
    d_out[OUT_SIZE + layer * HIDDEN + j]              = h_new;  // new_hidden
    d_out[OUT_SIZE + 2 * HIDDEN + layer * HIDDEN + j] = c_new;  // new_cell
}

// softmax over a size-1 axis == all ones, independent of logits.
__global__ void fill_ones(float* __restrict__ out, int n)
{
    int i = blockIdx.x * blockDim.x + threadIdx.x;
    if (i < n) out[i] = 1.0f;
}

extern "C" void kernel_launch(void* const* d_in, const int* in_sizes, int n_in,
                              void* d_out, int out_size, void* d_ws, size_t ws_size,
                              hipStream_t stream)
{
    (void)in_sizes; (void)n_in; (void)out_size; (void)ws_size;

    const float* x       = (const float*)d_in[0];
    const float* hidden  = (const float*)d_in[1];   // [2,1,256]
    const float* cell    = (const float*)d_in[2];   // [2,1,256]
    const float* in_w    = (const float*)d_in[3];   // [256,32768]
    const float* in_b    = (const float*)d_in[4];
    const float* prelu_w = (const float*)d_in[5];
    const float* w_ih_l0 = (const float*)d_in[6];
    const float* w_hh_l0 = (const float*)d_in[7];
    const float* b_ih_l0 = (const float*)d_in[8];
    const float* b_hh_l0 = (const float*)d_in[9];
    const float* w_ih_l1 = (const float*)d_in[10];
    const float* w_hh_l1 = (const float*)d_in[11];
    const float* b_ih_l1 = (const float*)d_in[12];
    const float* b_hh_l1 = (const float*)d_in[13];
    // d_in[14]=fc_w, d_in[15]=fc_b are dead: softmax(axis=0, size 1) == 1.0

    float* out = (float*)d_out;
    float* ws  = (float*)d_ws;

    // workspace layout (floats)
    float* pv  = ws;                // 64 * 256   in_layer partials
    float* v   = ws + 16384;        // 256        PReLU output
    float* pga = ws + 16640;        // 4 * 1024   w_ih partials
    float* pgb = ws + 20736;        // 4 * 1024   w_hh partials
    float* h0  = ws + 24832;        // 256
    float* h1  = ws + 25088;        // 256

    // probs = ones
    fill_ones<<<(OUT_SIZE + 255) / 256, 256, 0, stream>>>(out, OUT_SIZE);

    // v_pre = in_w @ x   (16 M-tiles x 64 K-slices of 512)
    wmma_matvec_partial<<<dim3(16, 64), 32, 0, stream>>>(in_w, x, pv,
                                                         HIDDEN, IN_SIZE, 512);
    reduce_bias_prelu<<<1, HIDDEN, 0, stream>>>(pv, in_b, prelu_w, v, HIDDEN, 64);

    // ---- LSTM layer 0 ----
    wmma_matvec_partial<<<dim3(64, 4), 32, 0, stream>>>(w_ih_l0, v,      pga,
                                                        4 * HIDDEN, HIDDEN, 64);
    wmma_matvec_partial<<<dim3(64, 4), 32, 0, stream>>>(w_hh_l0, hidden, pgb,
                                                        4 * HIDDEN, HIDDEN, 64);
    lstm_combine<<<1, HIDDEN, 0, stream>>>(pga, pgb, 4, b_ih_l0, b_hh_l0,
                                           cell, h0, out, 0);

    // ---- LSTM layer 1 ----
    wmma_matvec_partial<<<dim3(64, 4), 32, 0, stream>>>(w_ih_l1, h0,              pga,
                                                        4 * HIDDEN, HIDDEN, 64);
    wmma_matvec_partial<<<dim3(64, 4), 32, 0, stream>>>(w_hh_l1, hidden + HIDDEN, pgb,
                                                        4 * HIDDEN, HIDDEN, 64);
    lstm_combine<<<1, HIDDEN, 0, stream>>>(pga, pgb, 4, b_ih_l1, b_hh_l1,
                                           cell + HIDDEN, h1, out, 1);
}